// ManoBasic_19138374271589
// MI455X (gfx1250) — compile-verified
//
#include <hip/hip_runtime.h>
#include <hip/hip_bf16.h>

typedef __attribute__((ext_vector_type(2))) float v2f;
typedef __attribute__((ext_vector_type(8))) float v8f;

// ---------------- problem constants ----------------
#define BATCH   8192
#define NVERT   778
#define NJOINT  16
#define NPOSE   135
#define NFEAT   145          // 10 shape + 135 pose
#define KSTEPS  37           // ceil(148/4): K padded to 148
#define NVT     49           // vertex tiles of 16 (49*16 = 784 >= 778)
#define NRT     147          // row tiles of 16 (rows = NVERT*3 = 2334 -> 147*16 = 2352)
#define NBG     512          // batch groups of 16

#define ROWS    (NVERT*3)    // 2334
#define JOINT_BASE ((size_t)BATCH * (size_t)ROWS)   // start of joints in d_out

// workspace layout (in floats)
#define DPACK_OFF  0
#define N_DPACK    (NRT*KSTEPS*64)          // 348096
#define WPACK_OFF  (DPACK_OFF + N_DPACK)
#define N_WPACK    (NVT*4*64)               // 12544
#define JTMPL_OFF  (WPACK_OFF + N_WPACK)
#define N_JTMPL    (NJOINT*3)               // 48
#define JDIRS_OFF  (JTMPL_OFF + N_JTMPL)
#define N_JDIRS    (NJOINT*3*10)            // 480
#define MPACK_OFF  (JDIRS_OFF + N_JDIRS)
#define N_MPACK    (BATCH*4*64)             // 2097152

// ---------------------------------------------------------------------------
// Kernel P: prepack constants into WMMA operand layouts + reduce J_regressor.
//   Dpack[tile][s][lane][e] : A-operand of blend GEMM (rows x features)
//   Wpack[vt][s][lane][e]   : A-operand of skin GEMM  (verts x joints)
//   Jtmpl/Jdirs             : J_regressor @ template / @ shapedirs
// A layout (f32 16x16x4): lane&15 = M row, K = s*4 + 2*(lane>=16) + e
// ---------------------------------------------------------------------------
__global__ void mano_prepack_kernel(const float* __restrict__ tmpl,
                                    const float* __restrict__ shapedirs,
                                    const float* __restrict__ jreg,
                                    const float* __restrict__ posedirs,
                                    const float* __restrict__ lbs,
                                    float* __restrict__ ws)
{
    int idx = blockIdx.x * blockDim.x + threadIdx.x;

    if (idx < N_DPACK) {
        int e    = idx & 1;
        int lane = (idx >> 1) & 31;
        int s    = (idx >> 6) % KSTEPS;
        int tile = idx / (64 * KSTEPS);
        int row  = tile * 16 + (lane & 15);
        int k    = s * 4 + 2 * (lane >> 4) + e;
        float v = 0.0f;
        if (row < ROWS && k < NFEAT) {
            v = (k < 10) ? shapedirs[row * 10 + k]
                         : posedirs[(size_t)(k - 10) * ROWS + row];
        }
        ws[DPACK_OFF + idx] = v;
        return;
    }
    idx -= N_DPACK;
    if (idx < N_WPACK) {
        int e    = idx & 1;
        int lane = (idx >> 1) & 31;
        int s    = (idx >> 6) & 3;
        int vt   = idx >> 8;
        int v    = vt * 16 + (lane & 15);
        int k    = s * 4 + 2 * (lane >> 4) + e;   // joint index 0..15
        ws[WPACK_OFF + idx] = (v < NVERT) ? lbs[v * NJOINT + k] : 0.0f;
        return;
    }
    idx -= N_WPACK;
    if (idx < (NJOINT * 3 * 11)) {
        int l = idx % 11;            // 0 = template, 1..10 = shapedir l-1
        int k = (idx / 11) % 3;
        int j = idx / 33;
        float acc = 0.0f;
        for (int v = 0; v < NVERT; ++v) {
            float w = jreg[j * NVERT + v];
            float x = (l == 0) ? tmpl[v * 3 + k]
                               : shapedirs[(v * 3 + k) * 10 + (l - 1)];
            acc += w * x;
        }
        if (l == 0) ws[JTMPL_OFF + j * 3 + k] = acc;
        else        ws[JDIRS_OFF + (j * 3 + k) * 10 + (l - 1)] = acc;
    }
}

// ---------------------------------------------------------------------------
// Kernel 1: per-batch forward kinematics. One thread per batch.
// Emits skin matrices in the WMMA *B*-operand layout (K=joint, N=entry 0..15):
//   Mpack[b][s][lane][e] = Mskin[j = s*4 + 2*(lane>=16) + e][c = lane&15], c>=12 -> 0
// Also writes the 16 joints (rotated + translated, reordered) into d_out.
// ---------------------------------------------------------------------------
__device__ __forceinline__ void fk_emit(int b, int j,
                                        const float R[9], const float t[3],
                                        const float Jj[3],
                                        const float g[9], const float tr[3],
                                        float* __restrict__ ws,
                                        float* __restrict__ out)
{
    // corrected translation: t - R * J[j]
    float tc0 = t[0] - (R[0]*Jj[0] + R[1]*Jj[1] + R[2]*Jj[2]);
    float tc1 = t[1] - (R[3]*Jj[0] + R[4]*Jj[1] + R[5]*Jj[2]);
    float tc2 = t[2] - (R[6]*Jj[0] + R[7]*Jj[1] + R[8]*Jj[2]);
    float ent[12] = { R[0],R[1],R[2],tc0,  R[3],R[4],R[5],tc1,  R[6],R[7],R[8],tc2 };

    int s  = j >> 2;
    int hi = (j >> 1) & 1;
    int e  = j & 1;
    float* mp = ws + MPACK_OFF + ((size_t)b * 4 + s) * 64;
#pragma unroll
    for (int c = 0; c < 16; ++c)
        mp[(hi * 16 + c) * 2 + e] = (c < 12) ? ent[c] : 0.0f;

    // joint output: g * t + trans, scattered via ORDER-inverse
    const int dest16[16] = {0,5,6,7,9,10,11,17,18,19,13,14,15,1,2,3};
    size_t jo = JOINT_BASE + (size_t)b * 63 + (size_t)dest16[j] * 3;
    out[jo + 0] = g[0]*t[0] + g[1]*t[1] + g[2]*t[2] + tr[0];
    out[jo + 1] = g[3]*t[0] + g[4]*t[1] + g[5]*t[2] + tr[1];
    out[jo + 2] = g[6]*t[0] + g[7]*t[1] + g[8]*t[2] + tr[2];
}

__global__ void mano_fk_kernel(const float* __restrict__ grmt,
                               const float* __restrict__ finger,
                               const float* __restrict__ shape,
                               const float* __restrict__ trans,
                               float* __restrict__ ws,
                               float* __restrict__ out)
{
    int b = blockIdx.x * blockDim.x + threadIdx.x;
    if (b >= BATCH) return;

    float sh[10];
#pragma unroll
    for (int l = 0; l < 10; ++l) sh[l] = shape[b * 10 + l];

    // joints from reduced regressor
    float J[NJOINT][3];
#pragma unroll
    for (int j = 0; j < NJOINT; ++j)
#pragma unroll
        for (int k = 0; k < 3; ++k) {
            float a = ws[JTMPL_OFF + j * 3 + k];
            const float* d = ws + JDIRS_OFF + (j * 3 + k) * 10;
#pragma unroll
            for (int l = 0; l < 10; ++l) a += d[l] * sh[l];
            J[j][k] = a;
        }

    float g[9], tr[3];
#pragma unroll
    for (int i = 0; i < 9; ++i) g[i] = grmt[b * 9 + i];
#pragma unroll
    for (int i = 0; i < 3; ++i) tr[i] = trans[b * 3 + i];

    // root: R = I, t = J[0]
    const float I9[9] = {1,0,0, 0,1,0, 0,0,1};
    fk_emit(b, 0, I9, J[0], J[0], g, tr, ws, out);

    // five 3-joint chains, parents: (1+3f)->0, (2+3f)->(1+3f), (3+3f)->(2+3f)
    for (int f = 0; f < 5; ++f) {
        int j1 = 1 + 3 * f;
        float Rc[9], tc[3];
#pragma unroll
        for (int i = 0; i < 9; ++i) Rc[i] = finger[(size_t)b * 135 + (j1 - 1) * 9 + i];
#pragma unroll
        for (int k = 0; k < 3; ++k) tc[k] = J[j1][k];   // J0 + (J[j1]-J0)
        fk_emit(b, j1, Rc, tc, J[j1], g, tr, ws, out);

        for (int sj = 1; sj < 3; ++sj) {
            int j = j1 + sj;
            float Rl[9], Jrel[3], Rn[9], tn[3];
#pragma unroll
            for (int i = 0; i < 9; ++i) Rl[i] = finger[(size_t)b * 135 + (j - 1) * 9 + i];
#pragma unroll
            for (int k = 0; k < 3; ++k) Jrel[k] = J[j][k] - J[j - 1][k];
#pragma unroll
            for (int m = 0; m < 3; ++m) {
#pragma unroll
                for (int n = 0; n < 3; ++n)
                    Rn[m * 3 + n] = Rc[m*3+0]*Rl[0*3+n] + Rc[m*3+1]*Rl[1*3+n] + Rc[m*3+2]*Rl[2*3+n];
                tn[m] = Rc[m*3+0]*Jrel[0] + Rc[m*3+1]*Jrel[1] + Rc[m*3+2]*Jrel[2] + tc[m];
            }
            fk_emit(b, j, Rn, tn, J[j], g, tr, ws, out);
#pragma unroll
            for (int i = 0; i < 9; ++i) Rc[i] = Rn[i];
#pragma unroll
            for (int k = 0; k < 3; ++k) tc[k] = tn[k];
        }
    }
}

// ---------------------------------------------------------------------------
// Kernel 2: fused WMMA blend + skin. One wave (32 threads) per block.
// grid = (512 batch-groups, 49 vertex-tiles).
//   Phase 1: V' = Dpack x F  (3 row-tiles x 37 k-steps of v_wmma_f32_16x16x4_f32)
//   Phase 2: Mv = Wpack x Mpack (16 batches x 4 k-steps)
//   Phase 3: VALU apply (3x4 skin, 3x3 global rot, +trans), verts + fingertip joints.
// ---------------------------------------------------------------------------
__device__ __forceinline__ float featval(int k, int bglob,
                                         const float* __restrict__ shape,
                                         const float* __restrict__ finger)
{
    if (k < 10) return shape[bglob * 10 + k];
    if (k < NFEAT) {
        int p = k - 10;
        float v = finger[(size_t)bglob * 135 + p];
        int r = p % 9;
        if (r == 0 || r == 4 || r == 8) v -= 1.0f;   // pose_feature = R - I
        return v;
    }
    return 0.0f;
}

__launch_bounds__(32)
__global__ void mano_wmma_kernel(const float* __restrict__ grmt,
                                 const float* __restrict__ finger,
                                 const float* __restrict__ shape,
                                 const float* __restrict__ trans,
                                 const float* __restrict__ tmpl,
                                 const float* __restrict__ ws,
                                 float* __restrict__ out)
{
    __shared__ __attribute__((aligned(16))) float Fs[KSTEPS * 64];   // B-operand, blend GEMM
    __shared__ __attribute__((aligned(16))) float Vs[48 * 16];       // V' (rows x batch)
    __shared__ __attribute__((aligned(16))) float MvAll[16 * 16 * 16]; // [b][v][entry]
    __shared__ __attribute__((aligned(16))) float Gs[16 * 12];       // per-batch g(9)+trans(3)

    const int lane = threadIdx.x;
    const int bg   = blockIdx.x;
    const int vt   = blockIdx.y;
    const int hi   = lane >> 4;
    const int lo   = lane & 15;

    // ---- stage per-batch-group feature matrix into LDS (B-operand layout) ----
    {
        const int bglob = bg * 16 + lo;
        for (int s = 0; s < KSTEPS; ++s) {
            int kbase = s * 4 + 2 * hi;
            Fs[s * 64 + lane * 2 + 0] = featval(kbase + 0, bglob, shape, finger);
            Fs[s * 64 + lane * 2 + 1] = featval(kbase + 1, bglob, shape, finger);
        }
    }
    // ---- weight A-tile for this vertex tile: 8 VGPRs, reused for all batches ----
    v2f wA[4];
    {
        const float* wp = ws + WPACK_OFF + (size_t)(vt * 4) * 64;
#pragma unroll
        for (int s = 0; s < 4; ++s)
            wA[s] = *(const v2f*)(wp + s * 64 + lane * 2);
    }
    // ---- global rotations + translations ----
    if (lane < 16) {
        int bglob = bg * 16 + lane;
#pragma unroll
        for (int i = 0; i < 9; ++i) Gs[lane * 12 + i] = grmt[bglob * 9 + i];
#pragma unroll
        for (int i = 0; i < 3; ++i) Gs[lane * 12 + 9 + i] = trans[bglob * 3 + i];
    }
    __syncthreads();

    // ---- Phase 1: blend-shape GEMM, rows = 3 tiles of 16 ----
#pragma unroll
    for (int rt = 0; rt < 3; ++rt) {
        const int rtile = vt * 3 + rt;
        const float* dp = ws + DPACK_OFF + (size_t)rtile * KSTEPS * 64;
        v8f acc = {};
        for (int s = 0; s < KSTEPS; ++s) {
            v2f a = *(const v2f*)(dp + s * 64 + lane * 2);
            v2f bm = *(const v2f*)(Fs + s * 64 + lane * 2);
            acc = __builtin_amdgcn_wmma_f32_16x16x4_f32(false, a, false, bm,
                                                        (short)0, acc, false, false);
        }
        const int r0 = vt * 48 + rt * 16;
#pragma unroll
        for (int q = 0; q < 8; ++q) {
            int rl = q + 8 * hi;
            int r  = r0 + rl;
            float t = (r < ROWS) ? tmpl[r] : 0.0f;
            Vs[(rt * 16 + rl) * 16 + lo] = acc[q] + t;
        }
    }
    __syncthreads();

    // ---- Phase 2: skin-weight GEMM per batch ----
    for (int b = 0; b < 16; ++b) {
        const float* mp = ws + MPACK_OFF + ((size_t)(bg * 16 + b)) * 256;
        v8f mv = {};
#pragma unroll
        for (int s = 0; s < 4; ++s) {
            v2f bm = *(const v2f*)(mp + s * 64 + lane * 2);
            mv = __builtin_amdgcn_wmma_f32_16x16x4_f32(false, wA[s], false, bm,
                                                       (short)0, mv, false, false);
        }
#pragma unroll
        for (int q = 0; q < 8; ++q) {
            int vloc = q + 8 * hi;
            MvAll[(b * 16 + vloc) * 16 + lo] = mv[q];
        }
    }
    __syncthreads();

    // ---- Phase 3: apply skin matrix + global rotation + translation ----
    for (int iter = 0; iter < 8; ++iter) {
        int idx   = iter * 32 + lane;
        int b     = idx >> 4;
        int vloc  = idx & 15;
        int vglob = vt * 16 + vloc;
        if (vglob >= NVERT) continue;

        float px = Vs[(vloc * 3 + 0) * 16 + b];
        float py = Vs[(vloc * 3 + 1) * 16 + b];
        float pz = Vs[(vloc * 3 + 2) * 16 + b];

        const float* mr = &MvAll[(b * 16 + vloc) * 16];
        float sx = mr[0]*px + mr[1]*py + mr[2]*pz  + mr[3];
        float sy = mr[4]*px + mr[5]*py + mr[6]*pz  + mr[7];
        float sz = mr[8]*px + mr[9]*py + mr[10]*pz + mr[11];

        const float* gb = &Gs[b * 12];
        float vx = gb[0]*sx + gb[1]*sy + gb[2]*sz + gb[9];
        float vy = gb[3]*sx + gb[4]*sy + gb[5]*sz + gb[10];
        float vz = gb[6]*sx + gb[7]*sy + gb[8]*sz + gb[11];

        int bglob = bg * 16 + b;
        size_t o = (size_t)bglob * ROWS + (size_t)vglob * 3;
        out[o + 0] = vx; out[o + 1] = vy; out[o + 2] = vz;

        int tip = -1;
        if      (vglob == 745) tip = 0;
        else if (vglob == 317) tip = 1;
        else if (vglob == 444) tip = 2;
        else if (vglob == 556) tip = 3;
        else if (vglob == 673) tip = 4;
        if (tip >= 0) {
            const int tipdest[5] = {4, 8, 12, 16, 20};
            size_t jo = JOINT_BASE + (size_t)bglob * 63 + (size_t)tipdest[tip] * 3;
            out[jo + 0] = vx; out[jo + 1] = vy; out[jo + 2] = vz;
        }
    }
}

// ---------------------------------------------------------------------------
extern "C" void kernel_launch(void* const* d_in, const int* in_sizes, int n_in,
                              void* d_out, int out_size, void* d_ws, size_t ws_size,
                              hipStream_t stream)
{
    const float* grmt      = (const float*)d_in[0];  // (B,1,3,3)
    const float* finger    = (const float*)d_in[1];  // (B,15,3,3)
    const float* shape     = (const float*)d_in[2];  // (B,10)
    const float* trans     = (const float*)d_in[3];  // (B,3)
    const float* tmpl      = (const float*)d_in[4];  // (NV,3)
    const float* shapedirs = (const float*)d_in[5];  // (NV,3,10)
    const float* jreg      = (const float*)d_in[6];  // (16,NV)
    const float* posedirs  = (const float*)d_in[7];  // (135,NV*3)
    const float* lbs       = (const float*)d_in[8];  // (NV,16)
    (void)in_sizes; (void)n_in; (void)out_size; (void)ws_size;

    float* ws  = (float*)d_ws;
    float* out = (float*)d_out;

    // 1) prepack constant operands + reduced joint regressor
    {
        int total = N_DPACK + N_WPACK + NJOINT * 3 * 11;
        int blocks = (total + 255) / 256;
        mano_prepack_kernel<<<blocks, 256, 0, stream>>>(tmpl, shapedirs, jreg,
                                                        posedirs, lbs, ws);
    }
    // 2) per-batch forward kinematics -> skin matrices (WMMA B layout) + joints
    mano_fk_kernel<<<BATCH / 256, 256, 0, stream>>>(grmt, finger, shape, trans, ws, out);

    // 3) fused WMMA blend + skinning -> verts + fingertip joints
    {
        dim3 grid(NBG, NVT);
        mano_wmma_kernel<<<grid, 32, 0, stream>>>(grmt, finger, shape, trans,
                                                  tmpl, ws, out);
    }
}